// PixelContrastiveLoss_13383118094993
// MI455X (gfx1250) — compile-verified
//
#include <hip/hip_runtime.h>
#include <hip/hip_bf16.h>
#include <math.h>

typedef __attribute__((ext_vector_type(2))) float f32x2;
typedef __attribute__((ext_vector_type(8))) float f32x8;

#define C_DIM   256
#define WDIM    128
#define HWD     (128*128)
#define NNEG    256
#define NTILES  17            // ceil(257/16)
#define SIMS_STRIDE (NTILES*16)   // 272 sims slots per batch in workspace
#define TSTRIDE 260           // 260 % 64 = 4 -> conflict-free B-fragment reads
#define INV_T   (1.0f/0.07f)

// ---------------------------------------------------------------------------
// Kernel 1: one single-wave block per (batch, 16-vector tile).
// Gathers + computes 16 dot products via V_WMMA_F32_16X16X4_F32.
// grid = B * NTILES, block = 32 (one wave, EXEC all-ones at the WMMA).
// ---------------------------------------------------------------------------
__global__ __launch_bounds__(32)
void pcl_sims(const float* __restrict__ feat1,
              const float* __restrict__ feat2,
              const int*   __restrict__ pos_coords,
              const int*   __restrict__ neg_h,
              const int*   __restrict__ neg_w,
              float*       __restrict__ sims_ws)
{
    __shared__ float qlds[C_DIM];
    __shared__ float tbuf[16 * TSTRIDE];
    __shared__ float invn[16];

    const int b    = blockIdx.x / NTILES;
    const int t    = blockIdx.x % NTILES;
    const int lane = threadIdx.x;

    const int h1 = pos_coords[b * 4 + 0];
    const int w1 = pos_coords[b * 4 + 1];
    const int h2 = pos_coords[b * 4 + 2];
    const int w2 = pos_coords[b * 4 + 3];

    // ---- gather + L2-normalize q (8 channels per lane, wave-reduce norm) ----
    const float* qsrc = feat1 + (size_t)b * C_DIM * HWD + (size_t)h1 * WDIM + w1;
    float qv[8];
    float ssq = 0.f;
    #pragma unroll
    for (int j = 0; j < 8; ++j) {
        const int c = lane + 32 * j;
        qv[j] = qsrc[(size_t)c * HWD];      // 64KB-strided gather
        ssq += qv[j] * qv[j];
    }
    #pragma unroll
    for (int off = 16; off > 0; off >>= 1) ssq += __shfl_xor(ssq, off);
    const float qinv = 1.0f / fmaxf(sqrtf(ssq), 1e-12f);
    #pragma unroll
    for (int j = 0; j < 8; ++j) qlds[lane + 32 * j] = qv[j] * qinv;

    // ---- gather this block's 16 candidate vectors (2 lanes per vector) ----
    const int slot = lane >> 1;
    const int half = lane & 1;
    const int g    = t * 16 + slot;        // 0..255 = negs, 256 = positive, >256 pad
    float ss = 0.f;
    if (g <= NNEG) {
        int hh, ww;
        if (g < NNEG) { hh = neg_h[b * NNEG + g]; ww = neg_w[b * NNEG + g]; }
        else          { hh = h2;                  ww = w2; }
        const float* src = feat2 + (size_t)b * C_DIM * HWD + (size_t)hh * WDIM + ww;
        #pragma unroll 4
        for (int j = 0; j < 128; ++j) {
            const int c = half * 128 + j;
            float v = src[(size_t)c * HWD]; // 64KB-strided gather (latency bound)
            tbuf[slot * TSTRIDE + c] = v;
            ss += v * v;
        }
    } else {
        for (int j = 0; j < 128; ++j)
            tbuf[slot * TSTRIDE + half * 128 + j] = 0.f;
    }
    ss += __shfl_xor(ss, 1);               // pair-reduce squared norm
    if (half == 0) invn[slot] = 1.0f / fmaxf(sqrtf(ss), 1e-12f);

    // intra-wave LDS store->load fence before WMMA consumes q and the tile
    asm volatile("s_wait_dscnt 0" ::: "memory");

    // ---- f32 WMMA: D(16x16) += A(16x4, q replicated over M) * B(4x16, tile) ----
    const int   koff = (lane >> 4) * 2;               // hi half-wave holds K+2,K+3
    const float* brow = &tbuf[(lane & 15) * TSTRIDE]; // B column N = lane&15
    f32x8 acc = {};
    #pragma unroll 8
    for (int k = 0; k < C_DIM; k += 4) {
        f32x2 a;  a.x  = qlds[k + koff];  a.y  = qlds[k + koff + 1];
        f32x2 bb; bb.x = brow[k + koff];  bb.y = brow[k + koff + 1];
        acc = __builtin_amdgcn_wmma_f32_16x16x4_f32(
                  false, a, false, bb, (short)0, acc, false, false);
    }
    // row M=0 of D lives in acc[0] of lanes 0..15 (N = lane)
    if (lane < 16)
        sims_ws[b * SIMS_STRIDE + t * 16 + lane] = acc[0] * invn[lane];
}

// ---------------------------------------------------------------------------
// Kernel 2: per-batch 257-logit log-softmax loss (parallel reductions).
// grid = B, block = 64.
// ---------------------------------------------------------------------------
__global__ __launch_bounds__(64)
void pcl_loss(const float* __restrict__ sims_ws, float* __restrict__ loss_ws)
{
    __shared__ float red[64];
    __shared__ float bcast;
    const int b   = blockIdx.x;
    const int tid = threadIdx.x;
    const float* s = sims_ws + b * SIMS_STRIDE;

    float lmax = -INFINITY;
    for (int i = tid; i <= NNEG; i += 64) lmax = fmaxf(lmax, s[i] * INV_T);
    red[tid] = lmax;
    __syncthreads();
    if (tid == 0) {
        float m = -INFINITY;
        for (int i = 0; i < 64; ++i) m = fmaxf(m, red[i]);
        bcast = m;
    }
    __syncthreads();
    const float m = bcast;

    float lsum = 0.f;
    for (int i = tid; i <= NNEG; i += 64) lsum += expf(s[i] * INV_T - m);
    __syncthreads();
    red[tid] = lsum;
    __syncthreads();
    if (tid == 0) {
        float sum = 0.f;
        for (int i = 0; i < 64; ++i) sum += red[i];
        loss_ws[b] = -((s[NNEG] * INV_T) - (m + logf(sum)));
    }
}

// ---------------------------------------------------------------------------
// Kernel 3: deterministic batch mean (no float atomics).
// ---------------------------------------------------------------------------
__global__ void pcl_reduce(const float* __restrict__ loss_ws,
                           float* __restrict__ out, int nb)
{
    if (threadIdx.x == 0) {
        float s = 0.f;
        for (int i = 0; i < nb; ++i) s += loss_ws[i];
        out[0] = s / (float)nb;
    }
}

extern "C" void kernel_launch(void* const* d_in, const int* in_sizes, int n_in,
                              void* d_out, int out_size, void* d_ws, size_t ws_size,
                              hipStream_t stream)
{
    const float* feat1 = (const float*)d_in[0];
    const float* feat2 = (const float*)d_in[1];
    const int*   pos   = (const int*)d_in[2];
    const int*   nh    = (const int*)d_in[3];
    const int*   nw    = (const int*)d_in[4];

    const int B = in_sizes[2] / 4;            // pos_coords is (B,4)
    float* sims_ws = (float*)d_ws;            // B * 272 floats
    float* loss_ws = sims_ws + B * SIMS_STRIDE;  // B floats

    pcl_sims<<<B * NTILES, 32, 0, stream>>>(feat1, feat2, pos, nh, nw, sims_ws);
    pcl_loss<<<B, 64, 0, stream>>>(sims_ws, loss_ws);
    pcl_reduce<<<1, 32, 0, stream>>>(loss_ws, (float*)d_out, B);
}